// FlowPredictionModel_43344809952143
// MI455X (gfx1250) — compile-verified
//
#include <hip/hip_runtime.h>
#include <math.h>
#include <stdint.h>

// ---------------------------------------------------------------------------
// FlowPredictionModel for MI455X (gfx1250, wave32, WMMA, async-LDS DMA).
//
// Phase 1 (k_pre, fully parallel, bandwidth-bound): block s computes
//   z = [obs@W_obs+b_obs, ts] @ W_in + b_in for all 4096 rows, per-step batch
//   mean/var (BN1, biased) + ReLU -> u[s][b][4]; zeroes phase-2 sync slots.
//
// Phase 2 (k_scan, 1024 sequential steps, 16 co-resident WGs, 1 thread/row):
//   gates = [u|h] @ [Wih;Whh]^T via V_WMMA_F32_16X16X4_F32 (two accumulating
//   K=4 slices, bias folded into C), A staged in LDS in the ISA 16x4 layout.
//   u[s+1] is DMA'd into the next A-buffer with GLOBAL_LOAD_ASYNC_TO_LDS_B128
//   while step s computes (retired with s_wait_asynccnt). WMMA A/D tiles are
//   wave-local (32 rows/wave), so only wave_barriers guard the staging LDS.
//   BN2: wave butterfly -> LDS -> global atomicAdd into stats[s][8] -> split
//   barrier (cluster-barrier fast path + release/acquire arrive counter with
//   s_sleep backoff) -> normalize, y = BN2(h)@W_out + b_out.
//
// Workspace: u (S*B*4 f32) | stats (S*8 f32) | arrive (S u32)  = 64MB + 36KB.
// ---------------------------------------------------------------------------

#define S_LEN   1024
#define BATCH   4096
#define NBLK2   16
#define BN_EPS  1e-5f

typedef __attribute__((ext_vector_type(2))) float v2f;
typedef __attribute__((ext_vector_type(8))) float v8f;
typedef __attribute__((ext_vector_type(4))) int   v4i;

typedef __attribute__((address_space(1))) v4i gv4i;   // global int4
typedef __attribute__((address_space(3))) v4i lv4i;   // LDS int4

#if __has_builtin(__builtin_amdgcn_global_load_async_to_lds_b128) && \
    __has_builtin(__builtin_amdgcn_s_wait_asynccnt)
#define HAVE_ASYNC 1
#define WAIT_ASYNC(n) __builtin_amdgcn_s_wait_asynccnt(n)
#else
#define HAVE_ASYNC 0
#define WAIT_ASYNC(n)
#endif

// DMA 16B/lane from global into LDS (async, ASYNCcnt); sync fallback.
__device__ __forceinline__ void load_u_async(const float* g, float* l) {
#if HAVE_ASYNC
    // generic LDS address low 32 bits == LDS offset (ISA 10.2 aperture rule)
    __builtin_amdgcn_global_load_async_to_lds_b128(
        (gv4i*)(uintptr_t)g,
        (lv4i*)(unsigned)(uintptr_t)l,
        0, 0);
#else
    *(float4*)l = *(const float4*)g;
#endif
}

// Branch-free transcendentals: v_exp_f32 + v_rcp_f32, saturate correctly.
__device__ __forceinline__ float sigmoid_fast(float x) {
    return __builtin_amdgcn_rcpf(1.0f + __expf(-x));
}
__device__ __forceinline__ float tanh_fast(float x) {
    float e = __expf(2.0f * x);                       // +inf / 0 at extremes
    return 1.0f - 2.0f * __builtin_amdgcn_rcpf(e + 1.0f);
}

// ============================ Phase 1 ======================================
__global__ __launch_bounds__(256) void k_pre(
    const float* __restrict__ obs, const float* __restrict__ ts,
    const float* __restrict__ W_obs, const float* __restrict__ b_obs,
    const float* __restrict__ W_in,  const float* __restrict__ b_in,
    const float* __restrict__ bn1_g, const float* __restrict__ bn1_b,
    float* __restrict__ u, float* __restrict__ stats, unsigned int* __restrict__ arrive)
{
    __shared__ float wobs[64];
    __shared__ float win[48];
    __shared__ float cb[20];     // b_obs[8], b_in[4], bn1_g[4], bn1_b[4]
    __shared__ float red[8][8];
    __shared__ float bc[8];

    const int tid = threadIdx.x;
    const int s   = blockIdx.x;

    if (tid < 64)                 wobs[tid]          = W_obs[tid];
    else if (tid < 112)           win[tid - 64]      = W_in[tid - 64];
    else if (tid < 120)           cb[tid - 112]      = b_obs[tid - 112];
    else if (tid < 124)           cb[8 + tid - 120]  = b_in[tid - 120];
    else if (tid < 128)           cb[12 + tid - 124] = bn1_g[tid - 124];
    else if (tid < 132)           cb[16 + tid - 128] = bn1_b[tid - 128];
    if (tid < 8)  stats[s * 8 + tid] = 0.0f;     // reset phase-2 slots (ws poisoned)
    if (tid == 8) arrive[s] = 0u;
    __syncthreads();

    float z[16][4];
    float sm[4] = {0, 0, 0, 0}, sq[4] = {0, 0, 0, 0};

    #pragma unroll
    for (int i = 0; i < 16; ++i) {              // full unroll: z stays in VGPRs
        const int row = tid + i * 256;
        const float4* o4 = (const float4*)(obs + ((size_t)s * BATCH + row) * 8);
        float4 oa = o4[0], ob = o4[1];
        float  o[8] = {oa.x, oa.y, oa.z, oa.w, ob.x, ob.y, ob.z, ob.w};
        float4 t4 = *(const float4*)(ts + ((size_t)s * BATCH + row) * 4);
        float  tv[4] = {t4.x, t4.y, t4.z, t4.w};

        float feat[8];
        #pragma unroll
        for (int j = 0; j < 8; ++j) feat[j] = cb[j];
        #pragma unroll
        for (int k = 0; k < 8; ++k)
            #pragma unroll
            for (int j = 0; j < 8; ++j) feat[j] = fmaf(o[k], wobs[k * 8 + j], feat[j]);

        #pragma unroll
        for (int n = 0; n < 4; ++n) {
            float zz = cb[8 + n];
            #pragma unroll
            for (int j = 0; j < 8; ++j) zz = fmaf(feat[j], win[j * 4 + n], zz);
            #pragma unroll
            for (int j = 0; j < 4; ++j) zz = fmaf(tv[j], win[(8 + j) * 4 + n], zz);
            z[i][n] = zz;
            sm[n] += zz;
            sq[n] += zz * zz;
        }
    }

    #pragma unroll
    for (int m = 16; m >= 1; m >>= 1) {
        #pragma unroll
        for (int n = 0; n < 4; ++n) {
            sm[n] += __shfl_xor(sm[n], m, 32);
            sq[n] += __shfl_xor(sq[n], m, 32);
        }
    }
    const int wave = tid >> 5, lane = tid & 31;
    if (lane == 0) {
        #pragma unroll
        for (int n = 0; n < 4; ++n) { red[wave][n] = sm[n]; red[wave][4 + n] = sq[n]; }
    }
    __syncthreads();
    if (tid < 8) {
        float t = 0.0f;
        #pragma unroll
        for (int w = 0; w < 8; ++w) t += red[w][tid];
        bc[tid] = t;
    }
    __syncthreads();

    const float inv = 1.0f / (float)BATCH;
    float mean[4], rstd[4];
    #pragma unroll
    for (int n = 0; n < 4; ++n) {
        mean[n] = bc[n] * inv;
        float var = bc[4 + n] * inv - mean[n] * mean[n];
        rstd[n] = rsqrtf(var + BN_EPS);
    }

    #pragma unroll
    for (int i = 0; i < 16; ++i) {
        const int row = tid + i * 256;
        float4 uv;
        uv.x = fmaxf(fmaf(cb[12] * rstd[0], z[i][0] - mean[0], cb[16]), 0.0f);
        uv.y = fmaxf(fmaf(cb[13] * rstd[1], z[i][1] - mean[1], cb[17]), 0.0f);
        uv.z = fmaxf(fmaf(cb[14] * rstd[2], z[i][2] - mean[2], cb[18]), 0.0f);
        uv.w = fmaxf(fmaf(cb[15] * rstd[3], z[i][3] - mean[3], cb[19]), 0.0f);
        *(float4*)(u + ((size_t)s * BATCH + row) * 4) = uv;
    }
}

// ============================ Phase 2 ======================================
__global__ __launch_bounds__(256) void k_scan(
    const float* __restrict__ u,
    const float* __restrict__ W_ih, const float* __restrict__ W_hh,
    const float* __restrict__ b_ih, const float* __restrict__ b_hh,
    const float* __restrict__ bn2_g, const float* __restrict__ bn2_b,
    const float* __restrict__ W_out, const float* __restrict__ b_out,
    float* __restrict__ out, float* __restrict__ stats, unsigned int* __restrict__ arrive)
{
    __shared__ float Abuf[2][256 * 8];   // double-buffered [u(4)|h(4)] rows
    __shared__ float Gbuf[256 * 16];     // gates (WMMA D) staging
    __shared__ float red[8][8];

    const int tid  = threadIdx.x;
    const int b    = blockIdx.x * 256 + tid;
    const int lane = tid & 31, wave = tid >> 5;
    const int lh   = lane >> 4, lr = lane & 15;

    // Step-invariant WMMA B operands (K=4 x N=16 slices of [Wih;Whh]^T):
    // v0 holds K = 2*lh, v1 holds K = 2*lh+1, N = lr.
    v2f B0, B1;
    B0.x = W_ih[lr * 4 + 2 * lh];  B0.y = W_ih[lr * 4 + 2 * lh + 1];
    B1.x = W_hh[lr * 4 + 2 * lh];  B1.y = W_hh[lr * 4 + 2 * lh + 1];
    const float biasC = b_ih[lr] + b_hh[lr];      // C(M,N) = bias[N]

    float g2[4], bb2[4], wout[4][8], bo[8];
    #pragma unroll
    for (int n = 0; n < 4; ++n) { g2[n] = bn2_g[n]; bb2[n] = bn2_b[n]; }
    #pragma unroll
    for (int j = 0; j < 4; ++j)
        #pragma unroll
        for (int m = 0; m < 8; ++m) wout[j][m] = W_out[j * 8 + m];
    #pragma unroll
    for (int m = 0; m < 8; ++m) bo[m] = b_out[m];

    float h[4] = {0, 0, 0, 0}, c[4] = {0, 0, 0, 0};
    const float inv = 1.0f / (float)BATCH;

    // DMA u(s=0) into Abuf[0] while the setup above retires.
    load_u_async(u + (size_t)b * 4, &Abuf[0][tid * 8]);

    for (int s = 0; s < S_LEN; ++s) {
        const int cur = s & 1, nxt = cur ^ 1;

        // Issue next step's DMA, then wait for this step's buffer (in-order
        // async completion => <=1 outstanding retires the older transfer).
        if (s + 1 < S_LEN) {
            load_u_async(u + ((size_t)(s + 1) * BATCH + b) * 4, &Abuf[nxt][tid * 8]);
            if (s + 2 < S_LEN)   // warm L2 one step further out
                __builtin_prefetch(u + ((size_t)(s + 2) * BATCH + b) * 4, 0, 1);
        }
        float* ar = &Abuf[cur][tid * 8];
        ar[4] = h[0]; ar[5] = h[1]; ar[6] = h[2]; ar[7] = h[3];
        __builtin_amdgcn_wave_barrier();
        if (s + 1 < S_LEN) { WAIT_ASYNC(1); } else { WAIT_ASYNC(0); }

        // gates[32 rows/wave][16] = [u|h] @ [Wih;Whh]^T + bias via WMMA.
        // A and D tiles touch only this wave's 32 rows -> no block barrier.
        #pragma unroll
        for (int t = 0; t < 2; ++t) {
            const int base = wave * 32 + t * 16;
            const float* arow = &Abuf[cur][(base + lr) * 8];
            v2f a0, a1;                                   // ISA 16x4 A layout
            a0.x = arow[2 * lh];     a0.y = arow[2 * lh + 1];       // K-slice: u
            a1.x = arow[4 + 2 * lh]; a1.y = arow[4 + 2 * lh + 1];   // K-slice: h
            v8f acc;
            #pragma unroll
            for (int v = 0; v < 8; ++v) acc[v] = biasC;
            acc = __builtin_amdgcn_wmma_f32_16x16x4_f32(
                      false, a0, false, B0, (short)0, acc, false, false);
            acc = __builtin_amdgcn_wmma_f32_16x16x4_f32(
                      false, a1, false, B1, (short)0, acc, false, false);
            #pragma unroll
            for (int v = 0; v < 8; ++v)                   // D: row base+v+8*lh, col lr
                Gbuf[(base + v + 8 * lh) * 16 + lr] = acc[v];
        }
        __builtin_amdgcn_wave_barrier();

        float g[16];
        #pragma unroll
        for (int k = 0; k < 16; ++k) g[k] = Gbuf[tid * 16 + k];

        // LSTM pointwise (torch gate order i,f,g,o) — branch-free trans.
        #pragma unroll
        for (int j = 0; j < 4; ++j) {
            float ig = sigmoid_fast(g[j]);
            float fg = sigmoid_fast(g[4 + j]);
            float gg = tanh_fast(g[8 + j]);
            float og = sigmoid_fast(g[12 + j]);
            c[j] = fg * c[j] + ig * gg;
            h[j] = og * tanh_fast(c[j]);
        }

        // BN2 cross-batch reduction.
        float pv[8];
        #pragma unroll
        for (int j = 0; j < 4; ++j) { pv[j] = h[j]; pv[4 + j] = h[j] * h[j]; }
        #pragma unroll
        for (int m = 16; m >= 1; m >>= 1)
            #pragma unroll
            for (int k = 0; k < 8; ++k) pv[k] += __shfl_xor(pv[k], m, 32);
        if (lane == 0) {
            #pragma unroll
            for (int k = 0; k < 8; ++k) red[wave][k] = pv[k];
        }
        __syncthreads();
        if (tid < 8) {
            float t = 0.0f;
            #pragma unroll
            for (int w = 0; w < 8; ++w) t += red[w][tid];
            atomicAdd(stats + s * 8 + tid, t);
        }
        __threadfence();
        __syncthreads();

        // Split barrier across the 16 workgroups for step s.
        if (tid == 0) {
            __hip_atomic_fetch_add(arrive + s, 1u, __ATOMIC_ACQ_REL,
                                   __HIP_MEMORY_SCOPE_AGENT);
            while (__hip_atomic_load(arrive + s, __ATOMIC_ACQUIRE,
                                     __HIP_MEMORY_SCOPE_AGENT) < (unsigned)gridDim.x)
                __builtin_amdgcn_s_sleep(1);
        }
        __builtin_amdgcn_s_cluster_barrier();   // fast path if cluster-dispatched; NOP otherwise
        __syncthreads();
        __threadfence();

        float mean[4], rstd[4];
        #pragma unroll
        for (int j = 0; j < 4; ++j) {
            mean[j] = stats[s * 8 + j] * inv;
            float var = stats[s * 8 + 4 + j] * inv - mean[j] * mean[j];
            rstd[j] = rsqrtf(var + BN_EPS);
        }

        float y[8];
        #pragma unroll
        for (int m = 0; m < 8; ++m) y[m] = bo[m];
        #pragma unroll
        for (int j = 0; j < 4; ++j) {
            float hb = fmaf(g2[j] * rstd[j], h[j] - mean[j], bb2[j]);
            #pragma unroll
            for (int m = 0; m < 8; ++m) y[m] = fmaf(hb, wout[j][m], y[m]);
        }
        float* op = out + ((size_t)s * BATCH + b) * 8;
        *(float4*)(op)     = make_float4(y[0], y[1], y[2], y[3]);
        *(float4*)(op + 4) = make_float4(y[4], y[5], y[6], y[7]);
    }
}

// ============================ Launch =======================================
extern "C" void kernel_launch(void* const* d_in, const int* in_sizes, int n_in,
                              void* d_out, int out_size, void* d_ws, size_t ws_size,
                              hipStream_t stream)
{
    const float* obs   = (const float*)d_in[0];
    const float* ts    = (const float*)d_in[1];
    const float* W_obs = (const float*)d_in[2];
    const float* b_obs = (const float*)d_in[3];
    const float* W_in  = (const float*)d_in[4];
    const float* b_in  = (const float*)d_in[5];
    const float* bn1_g = (const float*)d_in[6];
    const float* bn1_b = (const float*)d_in[7];
    const float* W_ih  = (const float*)d_in[8];
    const float* W_hh  = (const float*)d_in[9];
    const float* b_ih  = (const float*)d_in[10];
    const float* b_hh  = (const float*)d_in[11];
    const float* bn2_g = (const float*)d_in[12];
    const float* bn2_b = (const float*)d_in[13];
    const float* W_out = (const float*)d_in[14];
    const float* b_out = (const float*)d_in[15];
    float* out = (float*)d_out;

    float* u = (float*)d_ws;
    float* stats = u + (size_t)S_LEN * BATCH * 4;
    unsigned int* arrive = (unsigned int*)(stats + (size_t)S_LEN * 8);

    k_pre<<<S_LEN, 256, 0, stream>>>(obs, ts, W_obs, b_obs, W_in, b_in,
                                     bn1_g, bn1_b, u, stats, arrive);
    k_scan<<<NBLK2, 256, 0, stream>>>(u, W_ih, W_hh, b_ih, b_hh,
                                      bn2_g, bn2_b, W_out, b_out,
                                      out, stats, arrive);
}